// VDP_Softmax_26499948216844
// MI455X (gfx1250) — compile-verified
//
#include <hip/hip_runtime.h>
#include <stdint.h>

// B=4096, C=128. out = [p (B*C) | sigma_y (B*C*C)] as f32.
// sigma_y = J S J^T with J = diag(p) - p p^T reduces to:
//   q = S^T p, u = S p, s = p.q
//   sigma_y[i,l] = p_i p_l S[i,l] - p_i*(p_l q_l) - (p_i u_i - s p_i)*p_l
// -> memory bound: 64KB in + 64KB out per batch. Load S once to LDS (async),
//    rank-1 corrections per 16x16 tile via one V_WMMA_F32_16X16X4_F32.

typedef __attribute__((ext_vector_type(2))) float v2f;
typedef __attribute__((ext_vector_type(8))) float v8f;

#define Csz   128
#define SROW  132   // padded row stride (floats): 16B-aligned, kills bank conflicts

__launch_bounds__(256)
__global__ void vdp_softmax_cov_kernel(const float* __restrict__ mu,
                                       const float* __restrict__ sigma,
                                       float* __restrict__ out,
                                       int nBatch) {
  extern __shared__ float smem[];
  float* sS  = smem;                 // [128][132]
  float* sP  = sS + Csz * SROW;      // [128]
  float* sU  = sP + Csz;             // [128]
  float* sQ  = sU + Csz;             // [128]
  float* red = sQ + Csz;             // [128]

  const int b   = blockIdx.x;
  const int tid = threadIdx.x;

  // ---- 1) async stream sigma[b] (128x128 f32) into LDS, 16B per lane ----
  {
    const uint64_t gbase  = (uint64_t)(uintptr_t)(sigma + (size_t)b * Csz * Csz);
    const uint32_t ldsTop = (uint32_t)(uintptr_t)(void*)sS;
    for (int t = tid; t < (Csz * Csz) / 4; t += 256) {
      const int e   = t << 2;                 // float index, multiple of 4
      const int row = e >> 7;
      const int col = e & (Csz - 1);
      const uint32_t ldsAddr = ldsTop + (uint32_t)(row * SROW + col) * 4u;
      const uint32_t goff    = (uint32_t)e * 4u;
      asm volatile("global_load_async_to_lds_b128 %0, %1, %2"
                   :: "v"(ldsAddr), "v"(goff), "s"(gbase) : "memory");
    }
  }

  // ---- 2) softmax over mu[b,:] (overlapped with async loads) ----
  float x = -3.402823466e38f;
  if (tid < Csz) { x = mu[(size_t)b * Csz + tid]; red[tid] = x; }
  __syncthreads();
  for (int st = 64; st; st >>= 1) {
    if (tid < st) red[tid] = fmaxf(red[tid], red[tid + st]);
    __syncthreads();
  }
  const float mx = red[0];
  __syncthreads();
  float ex = 0.0f;
  if (tid < Csz) { ex = __expf(x - mx); red[tid] = ex; }
  __syncthreads();
  for (int st = 64; st; st >>= 1) {
    if (tid < st) red[tid] += red[tid + st];
    __syncthreads();
  }
  const float inv = 1.0f / red[0];
  __syncthreads();
  if (tid < Csz) {
    const float pv = ex * inv;
    sP[tid] = pv;
    out[(size_t)b * Csz + tid] = pv;          // output #1: p
  }

  // ---- 3) wait for sigma in LDS, then u = S p (waves 0-3), q = S^T p (waves 4-7)
  asm volatile("s_wait_asynccnt 0x0" ::: "memory");
  __syncthreads();

  if (tid < Csz) {
    const int i = tid;
    float acc = 0.0f;
    #pragma unroll 4
    for (int k = 0; k < Csz; ++k) {
      const int kk = (k + i) & (Csz - 1);     // stagger: conflict-free row reads
      acc += sS[i * SROW + kk] * sP[kk];
    }
    sU[i] = acc;
  } else {
    const int l = tid - Csz;
    float acc = 0.0f;
    #pragma unroll 4
    for (int j = 0; j < Csz; ++j)             // column reads: bank = (l+4j)%64, clean
      acc += sP[j] * sS[j * SROW + l];
    sQ[l] = acc;
  }
  __syncthreads();

  // ---- 4) s = p . q ----
  if (tid < Csz) red[tid] = sP[tid] * sQ[tid];
  __syncthreads();
  for (int st = 64; st; st >>= 1) {
    if (tid < st) red[tid] += red[tid + st];
    __syncthreads();
  }
  const float sval = red[0];
  __syncthreads();

  // ---- 5) per-tile: D = A*B + C via V_WMMA_F32_16X16X4_F32 ----
  // A(16x4): col0 = p_M, col1 = p_M*u_M - s*p_M  (K=2..3 zero, lanes>=16)
  // B(4x16): row0 = -(p_N q_N), row1 = -p_N      (K=2..3 zero, lanes>=16)
  // C      : p_M * p_N * S[M,N]  (elementwise diag(p) S diag(p))
  const int lane  = tid & 31;
  const int w     = tid >> 5;      // wave id = output row-block (M/16)
  const int lh    = lane >> 4;     // lane half: A/B K-slots 2..3 (unused -> 0)
  const int ln    = lane & 15;
  const int Mbase = 16 * w;

  v2f Am;
  {
    const int M    = Mbase + ln;
    const float pm = sP[M];
    Am.x = lh ? 0.0f : pm;
    Am.y = lh ? 0.0f : (pm * sU[M] - sval * pm);
  }

  float* __restrict__ outSY =
      out + (size_t)nBatch * Csz + (size_t)b * Csz * Csz;

  #pragma unroll
  for (int c = 0; c < 8; ++c) {
    const int   N  = 16 * c + ln;
    const float pn = sP[N];
    v2f Bm;
    Bm.x = lh ? 0.0f : (-pn * sQ[N]);
    Bm.y = lh ? 0.0f : (-pn);

    v8f Cm;
    #pragma unroll
    for (int v = 0; v < 8; ++v) {
      const int M = Mbase + v + 8 * lh;       // C layout: vgpr v -> row M(+8 for hi half)
      Cm[v] = sP[M] * pn * sS[M * SROW + N];
    }

    v8f Dm = __builtin_amdgcn_wmma_f32_16x16x4_f32(
        /*neg_a=*/false, Am, /*neg_b=*/false, Bm,
        /*c_mod=*/(short)0, Cm, /*reuse_a=*/false, /*reuse_b=*/false);

    #pragma unroll
    for (int v = 0; v < 8; ++v) {
      const int M = Mbase + v + 8 * lh;
      outSY[M * Csz + N] = Dm[v];             // output #2: sigma_y
    }
  }
}

extern "C" void kernel_launch(void* const* d_in, const int* in_sizes, int n_in,
                              void* d_out, int out_size, void* d_ws, size_t ws_size,
                              hipStream_t stream) {
  const float* mu    = (const float*)d_in[0];
  const float* sigma = (const float*)d_in[1];
  float*       outp  = (float*)d_out;
  const int nBatch = in_sizes[0] / Csz;       // 4096
  const size_t shmem = (size_t)(Csz * SROW + 4 * Csz) * sizeof(float); // ~69.6 KB
  vdp_softmax_cov_kernel<<<dim3(nBatch), dim3(256), shmem, stream>>>(
      mu, sigma, outp, nBatch);
}